// SimpleDenguePredictor_20392504721599
// MI455X (gfx1250) — compile-verified
//
#include <hip/hip_runtime.h>
#include <hip/hip_bf16.h>
#include <math.h>

// Problem sizes (match reference)
#define N_MOL  131072
#define E_MOL  524288
#define G_SEG  4096
#define D_MOL  6
#define N_PROT 100000
#define E_PROT 1600000
#define D_PROT 20
#define H      64

typedef __attribute__((ext_vector_type(2))) float v2f;
typedef __attribute__((ext_vector_type(8))) float v8f;

// ---------------------------------------------------------------------------
// WMMA GEMM: out[nrows, 64] = act(in[nrows, K] @ W[Kw, 64] (+ bias))
//  * K is a compile-time constant, multiple of 4 (zero-padded weight rows
//    beyond Kw). nrows must be a multiple of 16 (true for all call sites).
//  * One wave owns a 16x16 output tile; 128-thread block (4 waves) covers all
//    64 output columns of a 16-row tile.
//  * W is staged in LDS as row-pairs so each B fragment is one ds_load_b64;
//    each A fragment is one contiguous global_load_b64.
//  * k-loop fully unrolled -> straight-line v_wmma_f32_16x16x4_f32 chain.
// ---------------------------------------------------------------------------
template <int K>
__global__ void wmma_gemm_t(const float* __restrict__ in,
                            const float* __restrict__ W, int Kw,
                            const float* __restrict__ bias,
                            float* __restrict__ out, int relu) {
    __shared__ v2f wl[(K / 2) * H];

    const int tid = threadIdx.x;
    // cooperative load of W (Kw x 64), zero-padded to K rows, as row pairs
    for (int p = tid; p < (K / 2) * H; p += 128) {
        const int kk = p >> 6;        // row-pair index
        const int c  = p & (H - 1);   // column
        v2f w;
        w.x = (2 * kk     < Kw) ? W[(2 * kk    ) * H + c] : 0.0f;
        w.y = (2 * kk + 1 < Kw) ? W[(2 * kk + 1) * H + c] : 0.0f;
        wl[p] = w;
    }
    __syncthreads();

    const int wave = tid >> 5;
    const int lane = tid & 31;
    const int half = lane >> 4;   // 0: K = {k,k+1}, 1: K = {k+2,k+3}
    const int idx  = lane & 15;
    const int col0 = wave * 16;
    const int row  = blockIdx.x * 16 + idx;

    const float* __restrict__ arow = in + (size_t)row * K;

    v8f acc = {};
#pragma unroll
    for (int k = 0; k < K; k += 4) {
        const int ka = k + 2 * half;
        v2f a = *(const v2f*)(arow + ka);              // global_load_b64
        v2f b = wl[(ka >> 1) * H + col0 + idx];        // ds_load_b64
        acc = __builtin_amdgcn_wmma_f32_16x16x4_f32(
            /*neg_a=*/false, a, /*neg_b=*/false, b,
            /*c_mod=*/(short)0, acc, /*reuse_a=*/false, /*reuse_b=*/false);
    }

    const float bv = bias ? bias[col0 + idx] : 0.0f;
    const int rbase = blockIdx.x * 16 + 8 * half;
#pragma unroll
    for (int i = 0; i < 8; ++i) {
        float v = acc[i] + bv;
        if (relu) v = fmaxf(v, 0.0f);
        out[(size_t)(rbase + i) * H + col0 + idx] = v;
    }
}

// zero-pad mol_x [N_MOL,6] -> [N_MOL,8]
__global__ void pad_molx_k(const float* __restrict__ x, float* __restrict__ xp) {
    int idx = blockIdx.x * blockDim.x + threadIdx.x;
    if (idx >= N_MOL * 8) return;
    int i = idx >> 3, j = idx & 7;
    xp[idx] = (j < D_MOL) ? x[i * D_MOL + j] : 0.0f;
}

// ---------------------------------------------------------------------------
// Small utility / streaming kernels
// ---------------------------------------------------------------------------
__global__ void fill_k(float* __restrict__ p, float v, int n) {
    int i = blockIdx.x * blockDim.x + threadIdx.x;
    if (i < n) p[i] = v;
}

__device__ __forceinline__ float lrelu(float x) { return x > 0.0f ? x : 0.2f * x; }

__device__ __forceinline__ void atomicMaxF(float* addr, float v) {
    if (v >= 0.0f) atomicMax((int*)addr, __float_as_int(v));
    else           atomicMin((unsigned int*)addr, __float_as_uint(v));
}

// ---- GCN ----
__global__ void gcn_deg_k(const int* __restrict__ dst, float* __restrict__ deg, int E) {
    int e = blockIdx.x * blockDim.x + threadIdx.x;
    if (e < E) atomicAdd(&deg[dst[e]], 1.0f);
}

__global__ void rsqrt_inplace_k(float* __restrict__ d, int n) {
    int i = blockIdx.x * blockDim.x + threadIdx.x;
    if (i < n) { float v = d[i]; d[i] = v > 0.0f ? rsqrtf(v) : 0.0f; }
}

// self-loop contribution: out[i,f] = h[i,f] * dinv[i]^2
__global__ void gcn_self_k(const float* __restrict__ h, const float* __restrict__ dinv,
                           float* __restrict__ out, int n) {
    int idx = blockIdx.x * blockDim.x + threadIdx.x;
    if (idx >= n * H) return;
    int i = idx >> 6;
    float di = dinv[i];
    out[idx] = h[idx] * di * di;
}

// edge scatter: out[dst,f] += h[src,f] * dinv[src]*dinv[dst]
__global__ void gcn_edge_k(const float* __restrict__ h,
                           const int* __restrict__ src, const int* __restrict__ dst,
                           const float* __restrict__ dinv, float* __restrict__ out, int E) {
    int idx = blockIdx.x * blockDim.x + threadIdx.x;
    if (idx >= E * H) return;
    int e = idx >> 6, f = idx & (H - 1);
    int s = src[e], d = dst[e];
    float c = dinv[s] * dinv[d];
    atomicAdd(&out[(long long)d * H + f], h[(long long)s * H + f] * c);
}

__global__ void bias_act_k(float* __restrict__ x, const float* __restrict__ b, int n, int relu) {
    int idx = blockIdx.x * blockDim.x + threadIdx.x;
    if (idx >= n * H) return;
    float v = x[idx] + b[idx & (H - 1)];
    if (relu) v = fmaxf(v, 0.0f);
    x[idx] = v;
}

// ---- GAT ----
__global__ void gat_scalars_k(const float* __restrict__ h, const float* __restrict__ as,
                              const float* __restrict__ ad, float* __restrict__ hs,
                              float* __restrict__ hd, int n) {
    int i = blockIdx.x * blockDim.x + threadIdx.x;
    if (i >= n) return;
    float s = 0.0f, d = 0.0f;
#pragma unroll 8
    for (int f = 0; f < H; ++f) {
        float v = h[(long long)i * H + f];
        s += v * as[f];
        d += v * ad[f];
    }
    hs[i] = s; hd[i] = d;
}

__global__ void gat_init_max_k(const float* __restrict__ hs, const float* __restrict__ hd,
                               float* __restrict__ mx, int n) {
    int i = blockIdx.x * blockDim.x + threadIdx.x;
    if (i < n) mx[i] = lrelu(hs[i] + hd[i]);   // self-loop edge
}

__global__ void gat_edge_max_k(const int* __restrict__ src, const int* __restrict__ dst,
                               const float* __restrict__ hs, const float* __restrict__ hd,
                               float* __restrict__ mx, int E) {
    int e = blockIdx.x * blockDim.x + threadIdx.x;
    if (e >= E) return;
    atomicMaxF(&mx[dst[e]], lrelu(hs[src[e]] + hd[dst[e]]));
}

__global__ void gat_init_sum_k(const float* __restrict__ hs, const float* __restrict__ hd,
                               const float* __restrict__ mx, float* __restrict__ sm, int n) {
    int i = blockIdx.x * blockDim.x + threadIdx.x;
    if (i < n) sm[i] = expf(lrelu(hs[i] + hd[i]) - mx[i]);
}

__global__ void gat_edge_sum_k(const int* __restrict__ src, const int* __restrict__ dst,
                               const float* __restrict__ hs, const float* __restrict__ hd,
                               const float* __restrict__ mx, float* __restrict__ sm, int E) {
    int e = blockIdx.x * blockDim.x + threadIdx.x;
    if (e >= E) return;
    int d = dst[e];
    atomicAdd(&sm[d], expf(lrelu(hs[src[e]] + hd[d]) - mx[d]));
}

// self-loop attention contribution
__global__ void gat_self_k(const float* __restrict__ h, const float* __restrict__ hs,
                           const float* __restrict__ hd, const float* __restrict__ mx,
                           const float* __restrict__ sm, float* __restrict__ out, int n) {
    int idx = blockIdx.x * blockDim.x + threadIdx.x;
    if (idx >= n * H) return;
    int i = idx >> 6;
    float alpha = expf(lrelu(hs[i] + hd[i]) - mx[i]) / sm[i];
    out[idx] = h[idx] * alpha;
}

__global__ void gat_edge_scatter_k(const float* __restrict__ h,
                                   const int* __restrict__ src, const int* __restrict__ dst,
                                   const float* __restrict__ hs, const float* __restrict__ hd,
                                   const float* __restrict__ mx, const float* __restrict__ sm,
                                   float* __restrict__ out, int E) {
    int idx = blockIdx.x * blockDim.x + threadIdx.x;
    if (idx >= E * H) return;
    int e = idx >> 6, f = idx & (H - 1);
    int s = src[e], d = dst[e];
    float alpha = expf(lrelu(hs[s] + hd[d]) - mx[d]) / sm[d];
    atomicAdd(&out[(long long)d * H + f], h[(long long)s * H + f] * alpha);
}

// ---- pooling / head ----
__global__ void pool_mol_k(const float* __restrict__ h, const int* __restrict__ batch,
                           float* __restrict__ sums, float* __restrict__ cnt) {
    int idx = blockIdx.x * blockDim.x + threadIdx.x;
    if (idx >= N_MOL * H) return;
    int i = idx >> 6, f = idx & (H - 1);
    int b = batch[i];
    atomicAdd(&sums[(long long)b * H + f], h[idx]);
    if (f == 0) atomicAdd(&cnt[b], 1.0f);
}

__global__ void prot_mean_k(const float* __restrict__ h, float* __restrict__ pmean, int n) {
    __shared__ float red[256];
    int f = blockIdx.x;                 // one block per feature (H blocks)
    float acc = 0.0f;
    for (int i = threadIdx.x; i < n; i += blockDim.x) acc += h[(long long)i * H + f];
    red[threadIdx.x] = acc;
    __syncthreads();
    for (int s = 128; s > 0; s >>= 1) {
        if ((int)threadIdx.x < s) red[threadIdx.x] += red[threadIdx.x + s];
        __syncthreads();
    }
    if (threadIdx.x == 0) pmean[f] = red[0] / (float)n;
}

__global__ void build_fused_k(const float* __restrict__ sums, const float* __restrict__ cnt,
                              const float* __restrict__ pmean, float* __restrict__ fused) {
    int idx = blockIdx.x * blockDim.x + threadIdx.x;
    if (idx >= G_SEG * H) return;
    int g = idx >> 6, f = idx & (H - 1);
    fused[g * (2 * H) + f]     = sums[idx] / fmaxf(cnt[g], 1.0f);
    fused[g * (2 * H) + H + f] = pmean[f];
}

__global__ void cls2_k(const float* __restrict__ z, const float* __restrict__ w,
                       const float* __restrict__ b, float* __restrict__ out) {
    int g = blockIdx.x * blockDim.x + threadIdx.x;
    if (g >= G_SEG) return;
    float s = b[0];
#pragma unroll 8
    for (int f = 0; f < H; ++f) s += z[g * H + f] * w[f];
    out[g] = 1.0f / (1.0f + expf(-s));
}

// ---------------------------------------------------------------------------
// Launch
// ---------------------------------------------------------------------------
static inline int ceil_div(long long a, int b) { return (int)((a + b - 1) / b); }

extern "C" void kernel_launch(void* const* d_in, const int* in_sizes, int n_in,
                              void* d_out, int out_size, void* d_ws, size_t ws_size,
                              hipStream_t stream) {
    const float* mol_x   = (const float*)d_in[0];
    const int*   mol_ei  = (const int*)  d_in[1];
    const int*   mol_b   = (const int*)  d_in[2];
    const float* prot_x  = (const float*)d_in[3];
    const int*   prot_ei = (const int*)  d_in[4];
    const float* gcn_w1  = (const float*)d_in[5];
    const float* gcn_b1  = (const float*)d_in[6];
    const float* gcn_w2  = (const float*)d_in[7];
    const float* gcn_b2  = (const float*)d_in[8];
    const float* gat_w1  = (const float*)d_in[9];
    const float* gat_as1 = (const float*)d_in[10];
    const float* gat_ad1 = (const float*)d_in[11];
    const float* gat_b1  = (const float*)d_in[12];
    const float* gat_w2  = (const float*)d_in[13];
    const float* gat_as2 = (const float*)d_in[14];
    const float* gat_ad2 = (const float*)d_in[15];
    const float* gat_b2  = (const float*)d_in[16];
    const float* cls_w1  = (const float*)d_in[17];
    const float* cls_b1  = (const float*)d_in[18];
    const float* cls_w2  = (const float*)d_in[19];
    const float* cls_b2  = (const float*)d_in[20];
    float* out = (float*)d_out;

    // workspace layout (floats)
    float* ws = (float*)d_ws;
    size_t o = 0;
    float* molx8  = ws + o; o += (size_t)N_MOL * 8;
    float* bufA_m = ws + o; o += (size_t)N_MOL * H;
    float* bufB_m = ws + o; o += (size_t)N_MOL * H;
    float* dinv_m = ws + o; o += (size_t)N_MOL;
    float* bufA_p = ws + o; o += (size_t)N_PROT * H;
    float* bufB_p = ws + o; o += (size_t)N_PROT * H;
    float* hs     = ws + o; o += (size_t)N_PROT;
    float* hd     = ws + o; o += (size_t)N_PROT;
    float* mx     = ws + o; o += (size_t)N_PROT;
    float* sm     = ws + o; o += (size_t)N_PROT;
    float* sums   = ws + o; o += (size_t)G_SEG * H;
    float* cnt    = ws + o; o += (size_t)G_SEG;
    float* pmean  = ws + o; o += (size_t)H;
    float* fused  = ws + o; o += (size_t)G_SEG * 2 * H;
    float* zbuf   = ws + o; o += (size_t)G_SEG * H;

    const int* m_src = mol_ei;
    const int* m_dst = mol_ei + E_MOL;
    const int* p_src = prot_ei;
    const int* p_dst = prot_ei + E_PROT;

    const int B = 256;

    // ======================= molecular GCN branch =======================
    // degrees with self-loops: deg = 1 + count(dst)
    fill_k<<<ceil_div(N_MOL, B), B, 0, stream>>>(dinv_m, 1.0f, N_MOL);
    gcn_deg_k<<<ceil_div(E_MOL, B), B, 0, stream>>>(m_dst, dinv_m, E_MOL);
    rsqrt_inplace_k<<<ceil_div(N_MOL, B), B, 0, stream>>>(dinv_m, N_MOL);

    // layer 1: pad K=6 -> 8, hW = x @ W1 (WMMA), propagate, +b, relu
    pad_molx_k<<<ceil_div((long long)N_MOL * 8, B), B, 0, stream>>>(mol_x, molx8);
    wmma_gemm_t<8><<<N_MOL / 16, 128, 0, stream>>>(molx8, gcn_w1, D_MOL, nullptr, bufA_m, 0);
    gcn_self_k<<<ceil_div((long long)N_MOL * H, B), B, 0, stream>>>(bufA_m, dinv_m, bufB_m, N_MOL);
    gcn_edge_k<<<ceil_div((long long)E_MOL * H, B), B, 0, stream>>>(bufA_m, m_src, m_dst, dinv_m, bufB_m, E_MOL);
    bias_act_k<<<ceil_div((long long)N_MOL * H, B), B, 0, stream>>>(bufB_m, gcn_b1, N_MOL, 1);

    // layer 2
    wmma_gemm_t<H><<<N_MOL / 16, 128, 0, stream>>>(bufB_m, gcn_w2, H, nullptr, bufA_m, 0);
    gcn_self_k<<<ceil_div((long long)N_MOL * H, B), B, 0, stream>>>(bufA_m, dinv_m, bufB_m, N_MOL);
    gcn_edge_k<<<ceil_div((long long)E_MOL * H, B), B, 0, stream>>>(bufA_m, m_src, m_dst, dinv_m, bufB_m, E_MOL);
    bias_act_k<<<ceil_div((long long)N_MOL * H, B), B, 0, stream>>>(bufB_m, gcn_b2, N_MOL, 0);

    // mean pool per graph
    fill_k<<<ceil_div(G_SEG * H, B), B, 0, stream>>>(sums, 0.0f, G_SEG * H);
    fill_k<<<ceil_div(G_SEG, B), B, 0, stream>>>(cnt, 0.0f, G_SEG);
    pool_mol_k<<<ceil_div((long long)N_MOL * H, B), B, 0, stream>>>(bufB_m, mol_b, sums, cnt);

    // ======================= protein GAT branch =======================
    // layer 1 (K=20 is a multiple of 4; N_PROT = 6250*16)
    wmma_gemm_t<D_PROT><<<N_PROT / 16, 128, 0, stream>>>(prot_x, gat_w1, D_PROT, nullptr, bufA_p, 0);
    gat_scalars_k<<<ceil_div(N_PROT, B), B, 0, stream>>>(bufA_p, gat_as1, gat_ad1, hs, hd, N_PROT);
    gat_init_max_k<<<ceil_div(N_PROT, B), B, 0, stream>>>(hs, hd, mx, N_PROT);
    gat_edge_max_k<<<ceil_div(E_PROT, B), B, 0, stream>>>(p_src, p_dst, hs, hd, mx, E_PROT);
    gat_init_sum_k<<<ceil_div(N_PROT, B), B, 0, stream>>>(hs, hd, mx, sm, N_PROT);
    gat_edge_sum_k<<<ceil_div(E_PROT, B), B, 0, stream>>>(p_src, p_dst, hs, hd, mx, sm, E_PROT);
    gat_self_k<<<ceil_div((long long)N_PROT * H, B), B, 0, stream>>>(bufA_p, hs, hd, mx, sm, bufB_p, N_PROT);
    gat_edge_scatter_k<<<ceil_div((long long)E_PROT * H, B), B, 0, stream>>>(bufA_p, p_src, p_dst, hs, hd, mx, sm, bufB_p, E_PROT);
    bias_act_k<<<ceil_div((long long)N_PROT * H, B), B, 0, stream>>>(bufB_p, gat_b1, N_PROT, 1);

    // layer 2
    wmma_gemm_t<H><<<N_PROT / 16, 128, 0, stream>>>(bufB_p, gat_w2, H, nullptr, bufA_p, 0);
    gat_scalars_k<<<ceil_div(N_PROT, B), B, 0, stream>>>(bufA_p, gat_as2, gat_ad2, hs, hd, N_PROT);
    gat_init_max_k<<<ceil_div(N_PROT, B), B, 0, stream>>>(hs, hd, mx, N_PROT);
    gat_edge_max_k<<<ceil_div(E_PROT, B), B, 0, stream>>>(p_src, p_dst, hs, hd, mx, E_PROT);
    gat_init_sum_k<<<ceil_div(N_PROT, B), B, 0, stream>>>(hs, hd, mx, sm, N_PROT);
    gat_edge_sum_k<<<ceil_div(E_PROT, B), B, 0, stream>>>(p_src, p_dst, hs, hd, mx, sm, E_PROT);
    gat_self_k<<<ceil_div((long long)N_PROT * H, B), B, 0, stream>>>(bufA_p, hs, hd, mx, sm, bufB_p, N_PROT);
    gat_edge_scatter_k<<<ceil_div((long long)E_PROT * H, B), B, 0, stream>>>(bufA_p, p_src, p_dst, hs, hd, mx, sm, bufB_p, E_PROT);
    bias_act_k<<<ceil_div((long long)N_PROT * H, B), B, 0, stream>>>(bufB_p, gat_b2, N_PROT, 0);

    prot_mean_k<<<H, 256, 0, stream>>>(bufB_p, pmean, N_PROT);

    // ======================= fuse + classifier =======================
    build_fused_k<<<ceil_div(G_SEG * H, B), B, 0, stream>>>(sums, cnt, pmean, fused);
    wmma_gemm_t<2 * H><<<G_SEG / 16, 128, 0, stream>>>(fused, cls_w1, 2 * H, cls_b1, zbuf, 1);
    cls2_k<<<ceil_div(G_SEG, B), B, 0, stream>>>(zbuf, cls_w2, cls_b2, out);
}